// GATModel_88905823027742
// MI455X (gfx1250) — compile-verified
//
#include <hip/hip_runtime.h>
#include <math.h>

// Model constants (match reference).
#define IN_DIM 128
#define HID    64
#define HEADS  4
#define HHID   256   // HEADS*HID
#define OUTD   2
#define NEG_SLOPE 0.2f
#define BN_EPS    1e-5f

typedef __attribute__((ext_vector_type(2))) float v2f;
typedef __attribute__((ext_vector_type(8))) float v8f;

// ---------------------------------------------------------------------------
// Monotone float <-> uint encoding so float segment-max can use atomicMax(u32).
// memset(0) on the encoded buffer is below every real value's encoding, so it
// doubles as the -inf sentinel (zero-in-degree nodes are never decoded).
// ---------------------------------------------------------------------------
__device__ __forceinline__ unsigned fenc(float f) {
  unsigned u = __float_as_uint(f);
  return (u & 0x80000000u) ? ~u : (u | 0x80000000u);
}
__device__ __forceinline__ float fdec(unsigned u) {
  return __uint_as_float((u & 0x80000000u) ? (u & 0x7FFFFFFFu) : ~u);
}

// ---------------------------------------------------------------------------
// K1: h1[N,256] = x[N,128] @ W1[128,256]  via V_WMMA_F32_16X16X4_F32.
// One wave computes one 16x16 output tile; 32 k-steps of K=4.
// Fragment layouts per CDNA5 ISA 7.12.2:
//   A 16x4 f32 : lanes 0-15 -> M=lane, {K=k0,k0+1}; lanes 16-31 -> {K=k0+2,k0+3}
//   B 4x16 f32 : VGPR0 = row k0 (lanes 0-15) / row k0+2 (lanes 16-31), VGPR1 = +1
//   C 16x16 f32: VGPR v = row v (lanes 0-15) / row v+8 (lanes 16-31), col = lane%16
// ---------------------------------------------------------------------------
#if __has_builtin(__builtin_amdgcn_wmma_f32_16x16x4_f32)
#define HAVE_WMMA_F32 1
__global__ __launch_bounds__(128) void gemm1_wmma(const float* __restrict__ x,
                                                  const float* __restrict__ W,
                                                  float* __restrict__ h1) {
  const int lane = threadIdx.x;          // wave32
  const int half = lane >> 4;            // 0 or 1
  const int l    = lane & 15;
  const int rowBase = blockIdx.x * 16;
  const int colBase = (blockIdx.y * 4 + threadIdx.y) * 16;   // 16 col tiles total

  const float* xrow = x + (size_t)(rowBase + l) * IN_DIM;
  v8f c = {0.f, 0.f, 0.f, 0.f, 0.f, 0.f, 0.f, 0.f};

#pragma unroll
  for (int k0 = 0; k0 < IN_DIM; k0 += 4) {
    const int ka = k0 + half * 2;
    v2f a, b;
    a.x = xrow[ka];
    a.y = xrow[ka + 1];
    b.x = W[(size_t)ka * HHID + colBase + l];
    b.y = W[(size_t)(ka + 1) * HHID + colBase + l];
    c = __builtin_amdgcn_wmma_f32_16x16x4_f32(false, a, false, b,
                                              (short)0, c, false, false);
  }

  float* out = h1 + (size_t)rowBase * HHID + colBase + l;
#pragma unroll
  for (int v = 0; v < 8; ++v)
    out[(size_t)(v + half * 8) * HHID] = c[v];
}
#endif

// Scalar GEMM for tail rows (N % 16) or as fallback if the builtin is absent.
__global__ void gemm1_plain(const float* __restrict__ x, const float* __restrict__ W,
                            float* __restrict__ h1, int row0, int nrows) {
  int idx = blockIdx.x * blockDim.x + threadIdx.x;
  if (idx >= nrows * HHID) return;
  int r = row0 + idx / HHID;
  int cidx = idx % HHID;
  float acc = 0.f;
  for (int k = 0; k < IN_DIM; ++k)
    acc += x[(size_t)r * IN_DIM + k] * W[(size_t)k * HHID + cidx];
  h1[(size_t)r * HHID + cidx] = acc;
}

// ---------------------------------------------------------------------------
// K2: per-(node,head) attention dots: a_src[n,h] = <h1[n,h,:], att_src[h,:]>.
// ---------------------------------------------------------------------------
__global__ void att1_kernel(const float* __restrict__ h1,
                            const float* __restrict__ att_s,
                            const float* __restrict__ att_d,
                            float* __restrict__ as, float* __restrict__ ad, int N) {
  int idx = blockIdx.x * blockDim.x + threadIdx.x;
  if (idx >= N * HEADS) return;
  int n = idx >> 2, h = idx & 3;
  const float* row = h1 + (size_t)n * HHID + h * HID;
  const float* wsv = att_s + h * HID;
  const float* wdv = att_d + h * HID;
  float s = 0.f, d = 0.f;
#pragma unroll 8
  for (int c = 0; c < HID; ++c) { float v = row[c]; s += v * wsv[c]; d += v * wdv[c]; }
  as[idx] = s;
  ad[idx] = d;
}

// ---------------------------------------------------------------------------
// K3/K4: layer-1 segment softmax over incoming edges (per edge-head element).
// ---------------------------------------------------------------------------
__global__ void edge1_max(const int* __restrict__ ei, const float* __restrict__ as,
                          const float* __restrict__ ad, float* __restrict__ ebuf,
                          unsigned* __restrict__ menc, int E) {
  int idx = blockIdx.x * blockDim.x + threadIdx.x;
  if (idx >= E * HEADS) return;
  int e = idx >> 2, h = idx & 3;
  int s = ei[e], d = ei[E + e];
  float v = as[s * HEADS + h] + ad[d * HEADS + h];
  v = v > 0.f ? v : NEG_SLOPE * v;                  // leaky_relu
  ebuf[idx] = v;
  atomicMax(&menc[d * HEADS + h], fenc(v));
}

__global__ void edge1_sum(const int* __restrict__ ei, const unsigned* __restrict__ menc,
                          float* __restrict__ ebuf, float* __restrict__ ssum, int E) {
  int idx = blockIdx.x * blockDim.x + threadIdx.x;
  if (idx >= E * HEADS) return;
  int e = idx >> 2, h = idx & 3;
  int d = ei[E + e];
  float m = fdec(menc[d * HEADS + h]);
  float p = expf(ebuf[idx] - m);
  ebuf[idx] = p;
  atomicAdd(&ssum[d * HEADS + h], p);
}

// K5: alpha-weighted scatter-add of 256 features, one block per edge.
// agg1 (102 MB) lives in the 192 MB L2, so the 410M atomics resolve on-chip.
__global__ __launch_bounds__(HHID) void edge1_agg(const int* __restrict__ ei,
                                                  const float* __restrict__ h1,
                                                  const float* __restrict__ ebuf,
                                                  const float* __restrict__ ssum,
                                                  float* __restrict__ agg, int E) {
  int e = blockIdx.x;
  int f = threadIdx.x;
  int s = ei[e], d = ei[E + e];
  int h = f >> 6;
  float alpha = ebuf[(size_t)e * HEADS + h] / (ssum[d * HEADS + h] + 1e-16f);
  atomicAdd(&agg[(size_t)d * HHID + f], h1[(size_t)s * HHID + f] * alpha);
}

// K6: bias + BatchNorm(eval) + ELU, elementwise. Writes hmid (reuses h1 buffer).
__global__ void bn_elu(const float* __restrict__ agg, const float* __restrict__ b1,
                       const float* __restrict__ gamma, const float* __restrict__ beta,
                       const float* __restrict__ rmean, const float* __restrict__ rvar,
                       float* __restrict__ hmid, int total) {
  int idx = blockIdx.x * blockDim.x + threadIdx.x;
  if (idx >= total) return;
  int f = idx & (HHID - 1);
  float v = agg[idx] + b1[f];
  v = (v - rmean[f]) * rsqrtf(rvar[f] + BN_EPS) * gamma[f] + beta[f];
  hmid[idx] = v > 0.f ? v : expm1f(v);
}

// K7: h2[N,2] = hmid[N,256] @ W2[256,2]  (wave-per-node, shfl reduction),
// fused with layer-2 attention dots.
__global__ __launch_bounds__(256) void gemm2_att(const float* __restrict__ hmid,
                                                 const float* __restrict__ W2,
                                                 const float* __restrict__ as2w,
                                                 const float* __restrict__ ad2w,
                                                 float* __restrict__ h2,
                                                 float* __restrict__ as2,
                                                 float* __restrict__ ad2, int N) {
  int wave = threadIdx.x >> 5;
  int lane = threadIdx.x & 31;
  int n = blockIdx.x * 8 + wave;
  if (n >= N) return;
  const float* row = hmid + (size_t)n * HHID;
  float a0 = 0.f, a1 = 0.f;
#pragma unroll
  for (int i = 0; i < HHID / 32; ++i) {
    float hv = row[lane + 32 * i];
    a0 += hv * W2[(lane + 32 * i) * OUTD + 0];
    a1 += hv * W2[(lane + 32 * i) * OUTD + 1];
  }
#pragma unroll
  for (int off = 16; off; off >>= 1) {
    a0 += __shfl_xor(a0, off, 32);
    a1 += __shfl_xor(a1, off, 32);
  }
  if (lane == 0) {
    h2[(size_t)n * OUTD + 0] = a0;
    h2[(size_t)n * OUTD + 1] = a1;
    as2[n] = a0 * as2w[0] + a1 * as2w[1];
    ad2[n] = a0 * ad2w[0] + a1 * ad2w[1];
  }
}

// K8/K9/K10: layer-2 segment softmax + aggregation (heads=1, C=2).
__global__ void edge2_max(const int* __restrict__ ei, const float* __restrict__ as,
                          const float* __restrict__ ad, float* __restrict__ ebuf,
                          unsigned* __restrict__ menc, int E) {
  int e = blockIdx.x * blockDim.x + threadIdx.x;
  if (e >= E) return;
  int s = ei[e], d = ei[E + e];
  float v = as[s] + ad[d];
  v = v > 0.f ? v : NEG_SLOPE * v;
  ebuf[e] = v;
  atomicMax(&menc[d], fenc(v));
}

__global__ void edge2_sum(const int* __restrict__ ei, const unsigned* __restrict__ menc,
                          float* __restrict__ ebuf, float* __restrict__ ssum, int E) {
  int e = blockIdx.x * blockDim.x + threadIdx.x;
  if (e >= E) return;
  int d = ei[E + e];
  float p = expf(ebuf[e] - fdec(menc[d]));
  ebuf[e] = p;
  atomicAdd(&ssum[d], p);
}

__global__ void edge2_agg(const int* __restrict__ ei, const float* __restrict__ h2,
                          const float* __restrict__ ebuf, const float* __restrict__ ssum,
                          float* __restrict__ agg2, int E) {
  int e = blockIdx.x * blockDim.x + threadIdx.x;
  if (e >= E) return;
  int s = ei[e], d = ei[E + e];
  float alpha = ebuf[e] / (ssum[d] + 1e-16f);
  atomicAdd(&agg2[(size_t)d * OUTD + 0], h2[(size_t)s * OUTD + 0] * alpha);
  atomicAdd(&agg2[(size_t)d * OUTD + 1], h2[(size_t)s * OUTD + 1] * alpha);
}

// K11: out = log_softmax(agg2 + b2).
__global__ void finalize(const float* __restrict__ agg2, const float* __restrict__ b2,
                         float* __restrict__ out, int N) {
  int n = blockIdx.x * blockDim.x + threadIdx.x;
  if (n >= N) return;
  float v0 = agg2[(size_t)n * OUTD + 0] + b2[0];
  float v1 = agg2[(size_t)n * OUTD + 1] + b2[1];
  float mx = fmaxf(v0, v1);
  float lse = logf(expf(v0 - mx) + expf(v1 - mx)) + mx;
  out[(size_t)n * OUTD + 0] = v0 - lse;
  out[(size_t)n * OUTD + 1] = v1 - lse;
}

// ---------------------------------------------------------------------------
extern "C" void kernel_launch(void* const* d_in, const int* in_sizes, int n_in,
                              void* d_out, int out_size, void* d_ws, size_t ws_size,
                              hipStream_t stream) {
  const float* x     = (const float*)d_in[0];
  const int*   ei    = (const int*)d_in[1];
  const float* W1    = (const float*)d_in[2];
  const float* as1w  = (const float*)d_in[3];
  const float* ad1w  = (const float*)d_in[4];
  const float* b1    = (const float*)d_in[5];
  const float* gamma = (const float*)d_in[6];
  const float* beta  = (const float*)d_in[7];
  const float* rmean = (const float*)d_in[8];
  const float* rvar  = (const float*)d_in[9];
  const float* W2    = (const float*)d_in[10];
  const float* as2w  = (const float*)d_in[11];
  const float* ad2w  = (const float*)d_in[12];
  const float* b2    = (const float*)d_in[13];
  float* out = (float*)d_out;

  const int N = in_sizes[0] / IN_DIM;
  const int E = in_sizes[1] / 2;

  // ---- workspace layout (floats). Zero-init region is contiguous & first. ----
  float* ws = (float*)d_ws;
  size_t off = 0;
  unsigned* menc1 = (unsigned*)(ws + off); off += (size_t)N * HEADS;
  float*    s1sum = ws + off;              off += (size_t)N * HEADS;
  unsigned* menc2 = (unsigned*)(ws + off); off += (size_t)N;
  float*    s2sum = ws + off;              off += (size_t)N;
  float*    agg1  = ws + off;              off += (size_t)N * HHID;
  float*    agg2  = ws + off;              off += (size_t)N * OUTD;
  const size_t zeroFloats = off;
  float* h1    = ws + off; off += (size_t)N * HHID;   // reused as hmid after BN
  float* ebuf1 = ws + off; off += (size_t)E * HEADS;
  float* asv1  = ws + off; off += (size_t)N * HEADS;
  float* adv1  = ws + off; off += (size_t)N * HEADS;
  float* h2    = ws + off; off += (size_t)N * OUTD;
  float* asv2  = ws + off; off += (size_t)N;
  float* adv2  = ws + off; off += (size_t)N;
  float* ebuf2 = ws + off; off += (size_t)E;
  (void)ws_size;

  hipMemsetAsync(d_ws, 0, zeroFloats * sizeof(float), stream);

  // K1: GEMM1 via WMMA (full 16-row tiles) + scalar tail.
  const int nFull = N / 16;
  const int tail  = N - nFull * 16;
#ifdef HAVE_WMMA_F32
  if (nFull > 0)
    gemm1_wmma<<<dim3(nFull, HHID / 64), dim3(32, 4), 0, stream>>>(x, W1, h1);
  if (tail > 0)
    gemm1_plain<<<(tail * HHID + 255) / 256, 256, 0, stream>>>(x, W1, h1, nFull * 16, tail);
#else
  gemm1_plain<<<((size_t)N * HHID + 255) / 256, 256, 0, stream>>>(x, W1, h1, 0, N);
#endif

  // K2: attention dots, layer 1.
  att1_kernel<<<(N * HEADS + 255) / 256, 256, 0, stream>>>(h1, as1w, ad1w, asv1, adv1, N);

  // K3-K5: layer-1 segment softmax + aggregation.
  edge1_max<<<(E * HEADS + 255) / 256, 256, 0, stream>>>(ei, asv1, adv1, ebuf1, menc1, E);
  edge1_sum<<<(E * HEADS + 255) / 256, 256, 0, stream>>>(ei, menc1, ebuf1, s1sum, E);
  edge1_agg<<<E, HHID, 0, stream>>>(ei, h1, ebuf1, s1sum, agg1, E);

  // K6: bias + BN + ELU -> hmid (reuse h1 buffer).
  bn_elu<<<((size_t)N * HHID + 255) / 256, 256, 0, stream>>>(agg1, b1, gamma, beta,
                                                             rmean, rvar, h1, N * HHID);

  // K7: GEMM2 + attention dots, layer 2.
  gemm2_att<<<(N + 7) / 8, 256, 0, stream>>>(h1, W2, as2w, ad2w, h2, asv2, adv2, N);

  // K8-K10: layer-2 segment softmax + aggregation.
  edge2_max<<<(E + 255) / 256, 256, 0, stream>>>(ei, asv2, adv2, ebuf2, menc2, E);
  edge2_sum<<<(E + 255) / 256, 256, 0, stream>>>(ei, menc2, ebuf2, s2sum, E);
  edge2_agg<<<(E + 255) / 256, 256, 0, stream>>>(ei, h2, ebuf2, s2sum, agg2, E);

  // K11: bias + log_softmax.
  finalize<<<(N + 255) / 256, 256, 0, stream>>>(agg2, b2, out, N);
}